// GNNEncoder_13469017440574
// MI455X (gfx1250) — compile-verified
//
#include <hip/hip_runtime.h>
#include <hip/hip_bf16.h>

typedef float v2f __attribute__((ext_vector_type(2)));
typedef float v8f __attribute__((ext_vector_type(8)));

#define FEAT 64

// ---------------- degree / normalization ----------------

__global__ __launch_bounds__(256) void k_init_deg(float* dis, int n) {
    int i = blockIdx.x * blockDim.x + threadIdx.x;
    if (i < n) dis[i] = 1.0f;               // self-loop contributes 1
}

__global__ __launch_bounds__(256) void k_scatter_deg(const int* __restrict__ ei,
                                                     float* dis, int nedges) {
    int e = blockIdx.x * blockDim.x + threadIdx.x;
    if (e < nedges) {
        int d = ei[(size_t)nedges + e];     // dst row of edge_index
        unsafeAtomicAdd(&dis[d], 1.0f);
    }
}

__global__ __launch_bounds__(256) void k_rsqrt(float* dis, int n) {
    int i = blockIdx.x * blockDim.x + threadIdx.x;
    if (i < n) dis[i] = rsqrtf(dis[i]);
}

// ---------------- WMMA GEMM: H = X(nrows x 64) @ W(64 x 64) ----------------
// One wave per 16-row tile; each wave produces all four 16x16 output tiles.
// fp32 matrix path: V_WMMA_F32_16X16X4_F32, 16 chained WMMAs per tile (K=64).
__global__ __launch_bounds__(256) void k_gemm_xw(const float* __restrict__ X,
                                                 const float* __restrict__ W,
                                                 float* __restrict__ Hout,
                                                 int nrows) {
    int gwave = (blockIdx.x * blockDim.x + threadIdx.x) >> 5;
    int ntile = (nrows + 15) >> 4;
    if (gwave >= ntile) return;             // uniform per-wave exit: EXEC stays all-ones
    int lane = threadIdx.x & 31;
    int half = lane >> 4;                   // 0: K=k,k+1   1: K=k+2,k+3
    int l15  = lane & 15;
    int rowBase = gwave << 4;

    // Preload all A fragments (16 k-steps, reused across the 4 column tiles).
    // A 16x4 fp32 layout: lanes 0-15 -> M=0..15 {K,K+1}; lanes 16-31 -> {K+2,K+3}.
    const float* xrow = X + (size_t)(rowBase + l15) * FEAT;
    v2f afrag[16];
#pragma unroll
    for (int kk = 0; kk < 16; ++kk) {
        int k = (kk << 2) + (half << 1);
        afrag[kk].x = xrow[k];
        afrag[kk].y = xrow[k + 1];
    }

#pragma unroll
    for (int c = 0; c < 4; ++c) {
        v8f acc = {0.f, 0.f, 0.f, 0.f, 0.f, 0.f, 0.f, 0.f};
        int colB = (c << 4) + l15;
#pragma unroll
        for (int kk = 0; kk < 16; ++kk) {
            int k = (kk << 2) + (half << 1);
            // B 4x16 fp32 layout mirrors A: lanes 0-15 -> N=0..15 {K,K+1};
            // lanes 16-31 -> {K+2,K+3}.
            v2f b;
            b.x = W[(size_t)k * FEAT + colB];
            b.y = W[(size_t)(k + 1) * FEAT + colB];
            acc = __builtin_amdgcn_wmma_f32_16x16x4_f32(
                false, afrag[kk], false, b, (short)0, acc, false, false);
        }
        // C/D layout: VGPR i -> lanes 0-15 {M=i, N=lane}, lanes 16-31 {M=i+8, N=lane-16}
        int rbase = rowBase + (half << 3);
#pragma unroll
        for (int i = 0; i < 8; ++i) {
            Hout[(size_t)(rbase + i) * FEAT + colB] = acc[i];
        }
    }
}

// ---------------- zero accumulator ----------------

__global__ __launch_bounds__(256) void k_zero(float4* buf, int nvec4) {
    int i = blockIdx.x * blockDim.x + threadIdx.x;
    if (i < nvec4) buf[i] = make_float4(0.f, 0.f, 0.f, 0.f);
}

// ---------------- edge scatter: agg[dst] += norm(e) * h[src] ----------------
// One wave per edge; 2 features per lane (64 features / 32 lanes).
// Feature matrices (25.6 MB) + edge index (12.8 MB) live in the 192 MB L2,
// so gathers and fp32 atomics resolve at L2 bandwidth, not HBM.
__global__ __launch_bounds__(256) void k_scatter_edges(const int* __restrict__ ei,
                                                       const float* __restrict__ dis,
                                                       const float* __restrict__ Hs,
                                                       float* __restrict__ agg,
                                                       int nedges) {
    int gid = blockIdx.x * blockDim.x + threadIdx.x;
    int e = gid >> 5;
    if (e >= nedges) return;
    int lane = gid & 31;
    int s = ei[e];
    int d = ei[(size_t)nedges + e];
    float w = dis[s] * dis[d];              // symmetric normalization
    const float2 v = *(const float2*)(Hs + (size_t)s * FEAT + lane * 2);
    float* out = agg + (size_t)d * FEAT + lane * 2;
    unsafeAtomicAdd(out,     v.x * w);
    unsafeAtomicAdd(out + 1, v.y * w);
}

// ---------------- finalize: y = [relu](agg + dis^2 * h + b), in place ----------------

__global__ __launch_bounds__(256) void k_finalize(float* __restrict__ agg,
                                                  const float* __restrict__ Hs,
                                                  const float* __restrict__ dis,
                                                  const float* __restrict__ bias,
                                                  int n, int do_relu) {
    int gid = blockIdx.x * blockDim.x + threadIdx.x;   // n*16 threads, 4 floats each
    int node = gid >> 4;
    if (node >= n) return;
    int f = (gid & 15) << 2;
    float sn = dis[node];
    sn *= sn;                                           // self-loop coefficient
    float4 a  = *(float4*)(agg + (size_t)node * FEAT + f);
    float4 hv = *(const float4*)(Hs + (size_t)node * FEAT + f);
    float4 bb = *(const float4*)(bias + f);
    a.x = fmaf(hv.x, sn, a.x) + bb.x;
    a.y = fmaf(hv.y, sn, a.y) + bb.y;
    a.z = fmaf(hv.z, sn, a.z) + bb.z;
    a.w = fmaf(hv.w, sn, a.w) + bb.w;
    if (do_relu) {
        a.x = fmaxf(a.x, 0.f); a.y = fmaxf(a.y, 0.f);
        a.z = fmaxf(a.z, 0.f); a.w = fmaxf(a.w, 0.f);
    }
    *(float4*)(agg + (size_t)node * FEAT + f) = a;
}

// ---------------- host launch ----------------

static inline int cdiv(long long a, long long b) { return (int)((a + b - 1) / b); }

extern "C" void kernel_launch(void* const* d_in, const int* in_sizes, int n_in,
                              void* d_out, int out_size, void* d_ws, size_t ws_size,
                              hipStream_t stream) {
    const float* x  = (const float*)d_in[0];
    const int*   ei = (const int*)d_in[1];          // int32 (JAX x64 disabled)
    const float* W1 = (const float*)d_in[3];
    const float* b1 = (const float*)d_in[4];
    const float* W2 = (const float*)d_in[5];
    const float* b2 = (const float*)d_in[6];
    const float* W3 = (const float*)d_in[7];
    const float* b3 = (const float*)d_in[8];
    float* out = (float*)d_out;

    const int n  = in_sizes[0] / FEAT;              // 100000
    const int ne = in_sizes[1] / 2;                 // 1600000

    // workspace layout: dis[n] | h[n*64] | A[n*64] | B[n*64]
    char* ws = (char*)d_ws;
    size_t off = (((size_t)n * 4) + 255) & ~(size_t)255;
    float* dis = (float*)ws;
    float* h   = (float*)(ws + off);  off += (size_t)n * FEAT * 4;
    float* bufA = (float*)(ws + off); off += (size_t)n * FEAT * 4;
    float* bufB = (float*)(ws + off);

    const int ntile = cdiv(n, 16);

    // 1) symmetric-norm coefficients
    k_init_deg<<<cdiv(n, 256), 256, 0, stream>>>(dis, n);
    k_scatter_deg<<<cdiv(ne, 256), 256, 0, stream>>>(ei, dis, ne);
    k_rsqrt<<<cdiv(n, 256), 256, 0, stream>>>(dis, n);

    // 2) three GCN layers
    struct { const float* in; const float* W; const float* b; float* agg; int relu; }
    layers[3] = {
        { x,    W1, b1, bufA, 1 },
        { bufA, W2, b2, bufB, 1 },
        { bufB, W3, b3, out,  0 },
    };
    for (int l = 0; l < 3; ++l) {
        k_gemm_xw<<<cdiv(ntile, 8), 256, 0, stream>>>(layers[l].in, layers[l].W, h, n);
        k_zero<<<cdiv((long long)n * 16, 256), 256, 0, stream>>>((float4*)layers[l].agg,
                                                                 n * 16);
        k_scatter_edges<<<cdiv((long long)ne * 32, 256), 256, 0, stream>>>(
            ei, dis, h, layers[l].agg, ne);
        k_finalize<<<cdiv((long long)n * 16, 256), 256, 0, stream>>>(
            layers[l].agg, h, dis, layers[l].b, n, layers[l].relu);
    }
}